// DeformableAttention_78288663872236
// MI455X (gfx1250) — compile-verified
//
#include <hip/hip_runtime.h>
#include <hip/hip_bf16.h>

#define NB   8
#define NC   128
#define NHH  96     // H
#define NWW  96     // W
#define HW   (NHH*NWW)
#define NHEADS 2
#define NPTS 8
#define HD   64
#define KCONV (NC*9)   // 1152

typedef __attribute__((ext_vector_type(2))) float v2f;
typedef __attribute__((ext_vector_type(8))) float v8f;

__device__ __forceinline__ v8f wmma_f32(v2f a, v2f b, v8f c) {
  // 8 args: (neg_a, A, neg_b, B, c_mod, C, reuse_a, reuse_b)
  return __builtin_amdgcn_wmma_f32_16x16x4_f32(
      false, a, false, b, (short)0, c, false, false);
}

// -------------------------------------------------------------------------
// Kernel 1: 3x3 conv (implicit GEMM via V_WMMA_F32_16X16X4_F32) + bias +
// per-head softmax over the 8 points, fused via cross-lane shuffles.
// Tap-major K ordering: k = tap*128 + c, so per-tap geometry/validity is
// hoisted out of the channel loop (no integer division in the hot loop).
// One wave computes a 16-pixel x 16-channel tile. 8 waves / block.
// -------------------------------------------------------------------------
__global__ __launch_bounds__(256) void conv_attn_softmax_k(
    const float* __restrict__ query, const float* __restrict__ attn_w,
    const float* __restrict__ attn_b, float* __restrict__ attn_out) {
  const int lane = threadIdx.x & 31;
  const int wave = threadIdx.x >> 5;
  const int tile = blockIdx.x * 8 + wave;
  const int p0 = tile * 16;                 // 16 consecutive pixels in one row
  const int b  = p0 / HW;
  int rem      = p0 - b * HW;
  const int y  = rem / NWW;
  const int x0 = rem - y * NWW;
  const int mn = lane & 15;                 // A: pixel row m; B: out-channel n
  const int kg = (lane >> 4) * 2;           // K sub-offset per ISA f32 A layout

  const float* qb = query + (size_t)b * NC * HW;

  v8f acc = {0.f,0.f,0.f,0.f,0.f,0.f,0.f,0.f};
#pragma unroll 1
  for (int tap = 0; tap < 9; ++tap) {
    const int dy = tap / 3 - 1;
    const int dx = tap - (tap / 3) * 3 - 1;
    const int yy = y + dy;
    const int xx = x0 + mn + dx;
    const bool valid = ((unsigned)yy < NHH) & ((unsigned)xx < NWW);
    const float vmask = valid ? 1.f : 0.f;
    const float* qtap = qb + (valid ? ((size_t)yy * NWW + xx) : (size_t)0);
    const float* wtap = attn_w + mn * KCONV + tap;   // + c*9 per channel
#pragma unroll 4
    for (int s = 0; s < NC / 4; ++s) {
      const int c = s * 4 + kg;
      v2f av, bv;
      av.x = qtap[(size_t)c * HW] * vmask;
      av.y = qtap[(size_t)(c + 1) * HW] * vmask;
      bv.x = wtap[c * 9];
      bv.y = wtap[(c + 1) * 9];
      acc = wmma_f32(av, bv, acc);
    }
  }

  // D layout: lane holds channel n=mn, rows m = 8*(lane>>4)+r in acc[r].
  // Softmax over points = over 8-lane groups (lanes [g*8, g*8+8)).
  const float bias = attn_b[mn];
#pragma unroll
  for (int r = 0; r < 8; ++r) {
    float v  = acc[r] + bias;
    float mx = v;
    mx = fmaxf(mx, __shfl_xor(mx, 1, 32));
    mx = fmaxf(mx, __shfl_xor(mx, 2, 32));
    mx = fmaxf(mx, __shfl_xor(mx, 4, 32));
    float e  = __expf(v - mx);
    float sm = e;
    sm += __shfl_xor(sm, 1, 32);
    sm += __shfl_xor(sm, 2, 32);
    sm += __shfl_xor(sm, 4, 32);
    const float a = e / sm;
    const int m = (lane >> 4) * 8 + r;
    attn_out[(((size_t)b * 16 + mn) * NHH + y) * NWW + (x0 + m)] = a;
  }
}

// -------------------------------------------------------------------------
// Kernel 2: bilinear grid-sample + attention-weighted accumulation.
// One thread per (b, head, y, x); gathers are L2-resident (2.25MB/slab).
// -------------------------------------------------------------------------
__global__ __launch_bounds__(256) void sample_weight_k(
    const float* __restrict__ value, const float* __restrict__ refpts,
    const float* __restrict__ attn, float* __restrict__ weighted) {
  const int t   = blockIdx.x * 256 + threadIdx.x;   // over B*NHEADS*H*W
  const int bn  = t / HW;
  int rem       = t - bn * HW;
  const int y   = rem / NWW;
  const int x   = rem - y * NWW;
  const int b   = bn >> 1;
  const int head = bn & 1;

  int   offs[NPTS][4];
  float ws[NPTS][4];
#pragma unroll
  for (int p = 0; p < NPTS; ++p) {
    const float ap = attn[(((size_t)b * 16 + head * NPTS + p) * NHH + y) * NWW + x];
    const size_t ri =
        (((((size_t)b * NHH + y) * NWW + x) * NHEADS + head) * NPTS + p) * 2;
    const float rx = refpts[ri + 0];
    const float ry = refpts[ri + 1];
    // grid = ref*2-1; x = ((g+1)*W - 1)*0.5  ==  ref*W - 0.5
    const float xf = rx * NWW - 0.5f;
    const float yf = ry * NHH - 0.5f;
    const float x0f = floorf(xf), y0f = floorf(yf);
    const float wx = xf - x0f, wy = yf - y0f;
    const int ix0 = (int)x0f, iy0 = (int)y0f;
    const int ix1 = ix0 + 1,  iy1 = iy0 + 1;
    const bool vx0 = (unsigned)ix0 < NWW, vx1 = (unsigned)ix1 < NWW;
    const bool vy0 = (unsigned)iy0 < NHH, vy1 = (unsigned)iy1 < NHH;
    const int cx0 = min(max(ix0, 0), NWW - 1), cx1 = min(max(ix1, 0), NWW - 1);
    const int cy0 = min(max(iy0, 0), NHH - 1), cy1 = min(max(iy1, 0), NHH - 1);
    offs[p][0] = cy0 * NWW + cx0;  offs[p][1] = cy0 * NWW + cx1;
    offs[p][2] = cy1 * NWW + cx0;  offs[p][3] = cy1 * NWW + cx1;
    ws[p][0] = ap * (1.f - wy) * (1.f - wx) * ((vy0 && vx0) ? 1.f : 0.f);
    ws[p][1] = ap * (1.f - wy) * wx         * ((vy0 && vx1) ? 1.f : 0.f);
    ws[p][2] = ap * wy * (1.f - wx)         * ((vy1 && vx0) ? 1.f : 0.f);
    ws[p][3] = ap * wy * wx                 * ((vy1 && vx1) ? 1.f : 0.f);
  }

  const float* vb = value    + ((size_t)b * NC + head * HD) * HW;
  float*       ob = weighted + ((size_t)b * NC + head * HD) * HW;
  const size_t opix = (size_t)y * NWW + x;
  for (int d = 0; d < HD; ++d) {
    const float* vd = vb + (size_t)d * HW;
    float a = 0.f;
#pragma unroll
    for (int p = 0; p < NPTS; ++p) {
      a += ws[p][0] * vd[offs[p][0]] + ws[p][1] * vd[offs[p][1]]
         + ws[p][2] * vd[offs[p][2]] + ws[p][3] * vd[offs[p][3]];
    }
    ob[(size_t)d * HW + opix] = a;
  }
}

// -------------------------------------------------------------------------
// Kernel 3: 1x1 projection as GEMM with V_WMMA_F32_16X16X4_F32.
// proj_w staged transposed in LDS [k][n] (64KB); the shared 16-pixel A
// tile staged once per block in LDS (8KB) with coalesced float4 loads so
// the 8 waves (one 16-wide N tile each) don't re-read global 8x.
// -------------------------------------------------------------------------
__global__ __launch_bounds__(256) void proj_gemm_k(
    const float* __restrict__ weighted, const float* __restrict__ proj_w,
    const float* __restrict__ proj_b, float* __restrict__ out) {
  __shared__ float sPW[NC * NC];   // [k][n]
  __shared__ float sA[NC * 16];    // [k][m]

  const int p0 = blockIdx.x * 16;
  const int b  = p0 / HW;
  int rem      = p0 - b * HW;
  const int y  = rem / NWW;
  const int x0 = rem - y * NWW;

  for (int i = threadIdx.x; i < NC * NC; i += 256) {
    const int k = i >> 7, n = i & 127;
    sPW[i] = proj_w[n * NC + k];
  }
  {
    const float* wbase = weighted + (size_t)b * NC * HW + (size_t)y * NWW + x0;
    for (int i = threadIdx.x; i < NC * 4; i += 256) {   // 512 float4 = 16*128
      const int k  = i >> 2;
      const int m4 = (i & 3) * 4;
      *reinterpret_cast<float4*>(&sA[k * 16 + m4]) =
          *reinterpret_cast<const float4*>(&wbase[(size_t)k * HW + m4]);
    }
  }
  __syncthreads();

  const int lane  = threadIdx.x & 31;
  const int ntile = threadIdx.x >> 5;      // 8 waves -> 8 N tiles
  const int mn = lane & 15;
  const int kg = (lane >> 4) * 2;
  const int n  = ntile * 16 + mn;

  v8f acc = {0.f,0.f,0.f,0.f,0.f,0.f,0.f,0.f};
#pragma unroll 4
  for (int s = 0; s < NC / 4; ++s) {
    const int k0 = s * 4 + kg;
    v2f av, bv;
    av.x = sA[(k0    ) * 16 + mn];
    av.y = sA[(k0 + 1) * 16 + mn];
    bv.x = sPW[(k0    ) * NC + n];
    bv.y = sPW[(k0 + 1) * NC + n];
    acc = wmma_f32(av, bv, acc);
  }

  const float bias = proj_b[n];
  const int m0 = (lane >> 4) * 8;
  float* op = out + (((size_t)b * NC + n) * NHH + y) * NWW + x0 + m0;
  float4 lo = make_float4(acc[0] + bias, acc[1] + bias, acc[2] + bias, acc[3] + bias);
  float4 hi = make_float4(acc[4] + bias, acc[5] + bias, acc[6] + bias, acc[7] + bias);
  *reinterpret_cast<float4*>(op)     = lo;
  *reinterpret_cast<float4*>(op + 4) = hi;
}

extern "C" void kernel_launch(void* const* d_in, const int* in_sizes, int n_in,
                              void* d_out, int out_size, void* d_ws, size_t ws_size,
                              hipStream_t stream) {
  const float* query  = (const float*)d_in[0];
  const float* value  = (const float*)d_in[1];
  const float* refpts = (const float*)d_in[2];
  const float* attn_w = (const float*)d_in[3];
  const float* attn_b = (const float*)d_in[4];
  const float* proj_w = (const float*)d_in[5];
  const float* proj_b = (const float*)d_in[6];
  float* out = (float*)d_out;

  float* attn     = (float*)d_ws;                  // B*16*H*W
  float* weighted = attn + (size_t)NB * 16 * HW;   // B*C*H*W

  // K1: 73728 px / (16 px/wave * 8 waves/block) = 576 blocks
  conv_attn_softmax_k<<<NB * HW / 128, 256, 0, stream>>>(
      query, attn_w, attn_b, attn);
  // K2: B*NHEADS*H*W / 256 = 576 blocks
  sample_weight_k<<<NB * NHEADS * HW / 256, 256, 0, stream>>>(
      value, refpts, attn, weighted);
  // K3: one 16-pixel tile per block = 4608 blocks
  proj_gemm_k<<<NB * HW / 16, 256, 0, stream>>>(
      weighted, proj_w, proj_b, out);
}